// RelativePositionalEmbedding2D_57887569215533
// MI455X (gfx1250) — compile-verified
//
#include <hip/hip_runtime.h>

typedef float v2f __attribute__((ext_vector_type(2)));
typedef float v4f __attribute__((ext_vector_type(4)));
typedef float v8f __attribute__((ext_vector_type(8)));

namespace {
constexpr int kTQ        = 4;
constexpr int kDim       = 64;    // reduction length (DIM)
constexpr int kRows      = 256;   // QH*QW rows per (b,head,t) slice
constexpr int kCols      = 1024;  // TK*KH*KW
constexpr int kColChunks = 8;     // column split for occupancy
constexpr int kColsPer   = kCols / kColChunks; // 128
constexpr int kThreads   = 512;   // 16 waves (wave32)
}

// One wave computes D(16x16) = A(16x64) * B(64x16) in fp32 via 16 chained
// V_WMMA_F32_16X16X4_F32. Per ISA 7.12.2, lane L supplies A row (L&15) and
// B row (L&15); VGPR pair {x,y} holds K offsets {2*(L>>4), 2*(L>>4)+1} of
// each 4-wide K step. bmask (0.0/1.0 per lane) zeroes B rows branch-free
// (used to pad rel_t's 4 valid columns); it is a literal 1.0f for the full
// tiles and folds away after inlining.
__device__ inline v8f wave_gemm_16x16x64(const float* __restrict__ arow,
                                         const float* __restrict__ brow,
                                         int khalf, float bmask)
{
    v8f acc = {};
#pragma unroll
    for (int k0 = 0; k0 < 16; ++k0) {
        const int coff = 4 * k0 + 2 * khalf;
        v2f a = *(const v2f*)(arow + coff);
        v2f b = *(const v2f*)(brow + coff);
        b.x *= bmask;
        b.y *= bmask;
        acc = __builtin_amdgcn_wmma_f32_16x16x4_f32(
            /*neg_a=*/false, a, /*neg_b=*/false, b,
            /*c_mod=*/(short)0, acc, /*reuse_a=*/false, /*reuse_b=*/false);
    }
    return acc;
}

__global__ __launch_bounds__(kThreads, 1)
void relpos_fused(const float* __restrict__ query,
                  const float* __restrict__ scores,
                  const float* __restrict__ hemb,   // [31,64]
                  const float* __restrict__ wemb,   // [31,64]
                  const float* __restrict__ temb,   // [7,64]
                  float* __restrict__ out)
{
    __shared__ float relH[kRows][16];
    __shared__ float relW[kRows][16];
    __shared__ float relT[kRows][4];

    const int slice  = blockIdx.x / kColChunks;   // (b*HEADS + n)*TQ + t
    const int cchunk = blockIdx.x % kColChunks;
    const int t      = slice % kTQ;

    const int tid   = threadIdx.x;
    const int wave  = tid >> 5;   // 0..15
    const int lane  = tid & 31;
    const int mlo   = lane & 15;  // A row / B row owned by this lane
    const int khalf = lane >> 4;  // which K pair of each 4-wide step

    const float* qs = query + (size_t)slice * kRows * kDim;

    // ---- Phase A: rel terms via fp32 WMMA, results to LDS ----
    // rel_h tile: h = wave. Rows m = w (contiguous q rows), B row n: hemb[h-n+15].
    {
        const float* arow = qs + (wave * 16 + mlo) * kDim;
        const float* brow = hemb + (wave - mlo + 15) * kDim;
        v8f d = wave_gemm_16x16x64(arow, brow, khalf, 1.0f);
#pragma unroll
        for (int j = 0; j < 8; ++j)
            relH[wave * 16 + j + 8 * khalf][mlo] = d[j];
    }
    // rel_w tile: w = wave. Rows m = h (stride-16 q rows), B row n: wemb[w-n+15].
    {
        const float* arow = qs + (mlo * 16 + wave) * kDim;
        const float* brow = wemb + (wave - mlo + 15) * kDim;
        v8f d = wave_gemm_16x16x64(arow, brow, khalf, 1.0f);
#pragma unroll
        for (int j = 0; j < 8; ++j)
            relW[(j + 8 * khalf) * 16 + wave][mlo] = d[j];
    }
    // rel_t: row chunk = wave (16 contiguous rows). Only B rows n<4 are real
    // (temb[t-n+3]); rows 4..15 read a clamped-valid row but are zeroed by
    // bmask, so only D cols 0..3 carry data.
    {
        const int ti = t - mlo + 3;
        const int tiClamped = ti < 0 ? 0 : ti;         // valid row in [0,6]
        const float bmask = (mlo < kTQ) ? 1.0f : 0.0f;
        const float* arow = qs + (wave * 16 + mlo) * kDim;
        const float* brow = temb + tiClamped * kDim;
        v8f d = wave_gemm_16x16x64(arow, brow, khalf, bmask);
        if (mlo < kTQ) {
#pragma unroll
            for (int j = 0; j < 8; ++j)
                relT[wave * 16 + j + 8 * khalf][mlo] = d[j];
        }
    }

    __syncthreads();

    // ---- Phase B: stream scores -> out (memory-bound part) ----
    const size_t base = (size_t)slice * kRows * kCols;
    const int col0 = cchunk * kColsPer;
#pragma unroll 4
    for (int i = tid; i < kRows * (kColsPer / 4); i += kThreads) {
        const int r   = i / (kColsPer / 4);
        const int c4  = i - r * (kColsPer / 4);
        const int col = col0 + c4 * 4;          // multiple of 4 -> same kh,tk
        const int tk  = col >> 8;
        const int kh  = (col >> 4) & 15;
        const int kw  = col & 15;
        const size_t off = base + (size_t)r * kCols + col;

        v4f sc = __builtin_nontemporal_load((const v4f*)(scores + off));
        const float add = relH[r][kh] + relT[r][tk];
        v4f o;
        o.x = sc.x + add + relW[r][kw + 0];
        o.y = sc.y + add + relW[r][kw + 1];
        o.z = sc.z + add + relW[r][kw + 2];
        o.w = sc.w + add + relW[r][kw + 3];
        __builtin_nontemporal_store(o, (v4f*)(out + off));
    }
}

extern "C" void kernel_launch(void* const* d_in, const int* in_sizes, int n_in,
                              void* d_out, int out_size, void* d_ws, size_t ws_size,
                              hipStream_t stream)
{
    const float* query  = (const float*)d_in[0];
    const float* scores = (const float*)d_in[1];
    const float* hemb   = (const float*)d_in[2];
    const float* wemb   = (const float*)d_in[3];
    const float* temb   = (const float*)d_in[4];
    float* out = (float*)d_out;

    const int nSlices = in_sizes[0] / (kRows * kDim); // B*HEADS*TQ = 128
    dim3 grid(nSlices * kColChunks), block(kThreads);
    relpos_fused<<<grid, block, 0, stream>>>(query, scores, hemb, wemb, temb, out);
}